// FilterInterpolationModule_35545149341804
// MI455X (gfx1250) — compile-verified
//
#include <hip/hip_runtime.h>
#include <stdint.h>

// FilterInterpolation (DAIN/MEMC-style) for MI455X gfx1250.
// Bandwidth-bound gather kernel: TDM streams filter weights (input3) into LDS
// asynchronously; interior fast path loads one 5x5 patch per pixel/channel.

#define FS 4
constexpr int B_  = 2;
constexpr int C_  = 3;
constexpr int H_  = 720;
constexpr int W_  = 1280;
constexpr int HW_ = H_ * W_;
constexpr int TPB = 256;   // 8 wave32 waves
constexpr int WV  = 32;

typedef unsigned int v4u __attribute__((ext_vector_type(4)));
typedef int          v4i __attribute__((ext_vector_type(4)));
typedef int          v8i __attribute__((ext_vector_type(8)));

// ---- Tensor Data Mover: async 2D tile load (global -> LDS) -----------------
// tile_d0 contiguous f32 along dim0, tile_d1 rows with stride_elems f32 stride.
__device__ __forceinline__ void tdm_load_2d_f32(uint32_t lds_addr, uint64_t gaddr,
                                                uint32_t tile_d0, uint32_t tile_d1,
                                                uint64_t stride_elems) {
  v4u g0;
  g0.x = 1u;                                               // count = 1 (valid D#)
  g0.y = lds_addr;                                         // LDS byte address
  g0.z = (uint32_t)(gaddr & 0xFFFFFFFFull);                // global_addr[31:0]
  g0.w = (uint32_t)((gaddr >> 32) & 0x1FFFFFFull)          // global_addr[56:32]
       | (2u << 30);                                       // type = 2 ("image")
  v8i g1;
  g1[0] = (int)(2u << 16);                                 // data_size = 2 (4 bytes)
  g1[1] = (int)((tile_d0 & 0xFFFFu) << 16);                // tensor_dim0[15:0]
  g1[2] = (int)(((tile_d0 >> 16) & 0xFFFFu)                // tensor_dim0[31:16]
              | ((tile_d1 & 0xFFFFu) << 16));              // tensor_dim1[15:0]
  g1[3] = (int)(((tile_d1 >> 16) & 0xFFFFu)                // tensor_dim1[31:16]
              | ((tile_d0 & 0xFFFFu) << 16));              // tile_dim0
  g1[4] = (int)(tile_d1 & 0xFFFFu);                        // tile_dim1 | tile_dim2=0
  g1[5] = (int)(stride_elems & 0xFFFFFFFFull);             // tensor_dim0_stride[31:0]
  g1[6] = (int)((stride_elems >> 32) & 0xFFFFull);         // stride[47:32] | dim1_stride lo = 0
  g1[7] = 0;
  v4i z4 = {0, 0, 0, 0};
#if defined(__clang_major__) && (__clang_major__ >= 23)
  v8i z8 = {0, 0, 0, 0, 0, 0, 0, 0};
  __builtin_amdgcn_tensor_load_to_lds(g0, g1, z4, z4, z8, 0);
#else
  __builtin_amdgcn_tensor_load_to_lds(g0, g1, z4, z4, 0);
#endif
}

__device__ __forceinline__ int clampi(int v, int lo, int hi) {
  return v < lo ? lo : (v > hi ? hi : v);
}

__global__ __launch_bounds__(TPB) void filter_interp_gfx1250(
    const float* __restrict__ in1,   // (B, C, H, W) image
    const float* __restrict__ in2,   // (B, 2, H, W) flow
    const float* __restrict__ in3,   // (B, 16, H, W) filter weights
    float* __restrict__ out) {       // (B, C, H, W)
  // LDS: per-wave slab of 16 weight planes x 32 lanes (16 KB total).
  __shared__ float wsm[8 * 16 * WV];

  const int tid  = threadIdx.x;
  const int lane = tid & (WV - 1);
  const unsigned wave = ((unsigned)__builtin_amdgcn_readfirstlane((int)tid)) >> 5;
  const int x0 = blockIdx.x * TPB;       // W = 1280 = 5 * 256, always full blocks
  const int y  = blockIdx.y;
  const int b  = blockIdx.z;
  const int x  = x0 + tid;

  // ---- Kick off async TDM: this wave's 16x32 f32 weight tile -> LDS --------
  const uint32_t lds_base = (uint32_t)(uintptr_t)(&wsm[0]) + wave * (16u * WV * 4u);
  const uint64_t g3 = (uint64_t)(uintptr_t)in3 +
      4ull * ((uint64_t)b * 16ull * (uint64_t)HW_ +
              (uint64_t)y * (uint64_t)W_ + (uint64_t)(x0 + (int)wave * WV));
  tdm_load_2d_f32(lds_base, g3, WV, 16u, (uint64_t)HW_);

  // ---- Overlap: flow load (NT: read-once stream) + address math ------------
  const size_t pix = (size_t)y * W_ + x;
  const float fxo = __builtin_nontemporal_load(&in2[((size_t)b * 2 + 0) * HW_ + pix]);
  const float fyo = __builtin_nontemporal_load(&in2[((size_t)b * 2 + 1) * HW_ + pix]);
  const float fx = (float)x + fxo;
  const float fy = (float)y + fyo;
  const float flx = floorf(fx);
  const float fly = floorf(fy);
  const int ixL = (int)flx;
  const int iyT = (int)fly;
  const float a  = fx - flx;
  const float bt = fy - fly;
  const bool valid = (ixL >= 0) && (ixL + 1 <= W_ - 1) && (iyT >= 0) && (iyT + 1 <= H_ - 1);

  // ---- Wait for TDM, read our 16 weights from LDS --------------------------
  __builtin_amdgcn_s_wait_tensorcnt(0);
  __asm__ volatile("" ::: "memory");   // keep ds reads below the wait

  float wk[16];
#pragma unroll
  for (int k = 0; k < 16; ++k)
    wk[k] = wsm[wave * (16u * WV) + (unsigned)k * WV + (unsigned)lane];

  if (!valid) {
#pragma unroll
    for (int c = 0; c < C_; ++c)
      __builtin_nontemporal_store(0.0f, &out[((size_t)(b * C_ + c)) * HW_ + pix]);
    return;
  }

  const float c00 = (1.0f - a) * (1.0f - bt);
  const float c10 = a * (1.0f - bt);
  const float c01 = (1.0f - a) * bt;
  const float c11 = a * bt;

  const bool interior = (ixL >= 1) && (ixL <= W_ - 4) && (iyT >= 1) && (iyT <= H_ - 4);

  if (interior) {
    // Fast path: TL/TR/BL/BR windows are unit shifts of one 5x5 patch.
    const float* base = in1 + (size_t)(b * C_) * HW_ +
                        (size_t)(iyT - 1) * W_ + (ixL - 1);
#pragma unroll
    for (int c = 0; c < C_; ++c) {
      const float* p = base + (size_t)c * HW_;
      float P[5][5];
#pragma unroll
      for (int j = 0; j < 5; ++j)
#pragma unroll
        for (int i = 0; i < 5; ++i)
          P[j][i] = p[(size_t)j * W_ + i];

      float TL = 0.f, TR = 0.f, BL = 0.f, BR = 0.f;
#pragma unroll
      for (int j = 0; j < 4; ++j)
#pragma unroll
        for (int i = 0; i < 4; ++i) {
          const float w = wk[j * 4 + i];
          TL = fmaf(w, P[j][i],         TL);
          TR = fmaf(w, P[j][i + 1],     TR);
          BL = fmaf(w, P[j + 1][i],     BL);
          BR = fmaf(w, P[j + 1][i + 1], BR);
        }
      const float v = fmaf(c00, TL, fmaf(c10, TR, fmaf(c01, BL, c11 * BR)));
      __builtin_nontemporal_store(v, &out[((size_t)(b * C_ + c)) * HW_ + pix]);
    }
  } else {
    // Border path (<1% of pixels): exact reference double-clipping.
    // Keep it rolled across channels to limit I-cache footprint.
#pragma unroll 1
    for (int c = 0; c < C_; ++c) {
      const float* p = in1 + (size_t)(b * C_ + c) * HW_;
      float TL = 0.f, TR = 0.f, BL = 0.f, BR = 0.f;
#pragma unroll
      for (int dj = -1; dj <= 2; ++dj) {
        const int yT = clampi(iyT + dj,     0, H_ - 1);
        const int yB = clampi(iyT + 1 + dj, 0, H_ - 1);
#pragma unroll
        for (int di = -1; di <= 2; ++di) {
          const float w = wk[(dj + 1) * 4 + (di + 1)];
          const int xL = clampi(ixL + di,     0, W_ - 1);
          const int xR = clampi(ixL + 1 + di, 0, W_ - 1);
          TL = fmaf(w, p[(size_t)yT * W_ + xL], TL);
          TR = fmaf(w, p[(size_t)yT * W_ + xR], TR);
          BL = fmaf(w, p[(size_t)yB * W_ + xL], BL);
          BR = fmaf(w, p[(size_t)yB * W_ + xR], BR);
        }
      }
      const float v = fmaf(c00, TL, fmaf(c10, TR, fmaf(c01, BL, c11 * BR)));
      __builtin_nontemporal_store(v, &out[((size_t)(b * C_ + c)) * HW_ + pix]);
    }
  }
}

extern "C" void kernel_launch(void* const* d_in, const int* in_sizes, int n_in,
                              void* d_out, int out_size, void* d_ws, size_t ws_size,
                              hipStream_t stream) {
  (void)in_sizes; (void)n_in; (void)out_size; (void)d_ws; (void)ws_size;
  const float* in1 = (const float*)d_in[0];
  const float* in2 = (const float*)d_in[1];
  const float* in3 = (const float*)d_in[2];
  float* out = (float*)d_out;
  dim3 grid(W_ / TPB, H_, B_);   // 5 x 720 x 2 blocks, 256 threads each
  filter_interp_gfx1250<<<grid, TPB, 0, stream>>>(in1, in2, in3, out);
}